// DiscriminativeLoss_11759620456905
// MI455X (gfx1250) — compile-verified
//
#include <hip/hip_runtime.h>
#include <math.h>

typedef float v2f __attribute__((ext_vector_type(2)));
typedef float v8f __attribute__((ext_vector_type(8)));

#define B_ 4
#define C_ 16
#define K_ 16
#define HW_ (512 * 512)

#define DELTA_PULL 0.5f
#define DELTA_PUSH 1.5f
#define EPS 1e-6f

#define THREADS 256
#define BLOCKS_PER_B 64
#define WAVES_PER_B (BLOCKS_PER_B * (THREADS / 32)) /* 512 */
#define PIX_PER_WAVE (HW_ / WAVES_PER_B)            /* 512 */

// ---------------------------------------------------------------------------
// Zero the accumulator workspace (sums[B][K][C], sq_sums[B][K][C], cnts[B][K])
// ---------------------------------------------------------------------------
__global__ void zero_ws_kernel(float* __restrict__ ws, int n) {
    int i = blockIdx.x * blockDim.x + threadIdx.x;
    if (i < n) ws[i] = 0.0f;
}

// ---------------------------------------------------------------------------
// Heavy kernel: per-wave WMMA reduction over a pixel chunk.
//   D[c][k] += sum_p emb[b][c][p] * mask[b][k][p]      (and emb^2 variant)
// A tile (16x4 f32): lane m=l&15 -> channel row; half=l>>4 selects pixel pair.
// B tile (4x16 f32): lane m=l&15 -> instance col; same pixel-pair striping.
// C/D tile: VGPR j, lanes 0-15 -> (c=j, k=lane); lanes 16-31 -> (c=j+8, k=lane-16).
// ---------------------------------------------------------------------------
__global__ __launch_bounds__(THREADS) void seg_reduce_kernel(
    const float* __restrict__ emb, const int* __restrict__ msk,
    float* __restrict__ sums, float* __restrict__ sqs, float* __restrict__ cnts) {
    const int gtid = blockIdx.x * THREADS + (int)threadIdx.x;
    const int wave = gtid >> 5;
    const int lane = threadIdx.x & 31;
    const int b = wave / WAVES_PER_B;
    const int wc = wave % WAVES_PER_B;
    const int m = lane & 15;     // channel row (A) / instance col (B)
    const int half = lane >> 4;  // pixel-pair select within the K=4 slice

    const float* embR = emb + (size_t)b * C_ * HW_ + (size_t)m * HW_;
    const int* mskR = msk + (size_t)b * K_ * HW_ + (size_t)m * HW_;

    int p = wc * PIX_PER_WAVE + 2 * half;

    v8f accS = {};  // sums tile
    v8f accQ = {};  // sq_sums tile
    float lcnt = 0.0f;

#pragma unroll 4
    for (int it = 0; it < PIX_PER_WAVE / 4; ++it, p += 4) {
        v2f a = *(const v2f*)(embR + p);      // 2 consecutive pixels, one channel
        int2 mi = *(const int2*)(mskR + p);   // 2 consecutive pixels, one instance
        v2f bm;
        bm.x = (mi.x > 0) ? 1.0f : 0.0f;
        bm.y = (mi.y > 0) ? 1.0f : 0.0f;
        lcnt += bm.x + bm.y;
        v2f a2 = a * a;
        accS = __builtin_amdgcn_wmma_f32_16x16x4_f32(
            /*neg_a=*/false, a, /*neg_b=*/false, bm, (short)0, accS, false, false);
        accQ = __builtin_amdgcn_wmma_f32_16x16x4_f32(
            /*neg_a=*/false, a2, /*neg_b=*/false, bm, (short)0, accQ, false, false);
    }

    // Write back partial tiles: D element (c = j + 8*half, k = m)
    float* sB = sums + ((size_t)b * K_ + m) * C_;
    float* qB = sqs + ((size_t)b * K_ + m) * C_;
#pragma unroll
    for (int j = 0; j < 8; ++j) {
        const int c = j + 8 * half;
        unsafeAtomicAdd(&sB[c], accS[j]);
        unsafeAtomicAdd(&qB[c], accQ[j]);
    }
    unsafeAtomicAdd(&cnts[b * K_ + m], lcnt);
}

// ---------------------------------------------------------------------------
// Tiny epilogue: pull/push from the [B,K,C] statistics, single 64-thread block.
// ---------------------------------------------------------------------------
__global__ __launch_bounds__(64) void finalize_kernel(
    const float* __restrict__ sums, const float* __restrict__ sqs,
    const float* __restrict__ cnts, float* __restrict__ out) {
    __shared__ float sMean[B_ * K_][C_];
    __shared__ float sPull[B_ * K_];
    __shared__ float sValid[B_ * K_];
    __shared__ float sLoss[B_];

    const int t = threadIdx.x;  // 0..63 == b*16 + k
    const int b = t >> 4;
    {
        const float cnt = cnts[t];
        const bool valid = cnt > 0.0f;
        const float safe = fmaxf(cnt, 1.0f);
        float S2 = 0.0f, Sq = 0.0f;
        for (int c = 0; c < C_; ++c) {
            const float s = sums[t * C_ + c];
            sMean[t][c] = s / safe;
            S2 += s * s;
            Sq += sqs[t * C_ + c];
        }
        // sum_c ||e - mean||^2 over fg pixels == Sq - S2/cnt   (when valid)
        sPull[t] = valid ? (Sq - S2 / cnt) / (cnt + EPS) : 0.0f;
        sValid[t] = valid ? 1.0f : 0.0f;
    }
    __syncthreads();

    if ((t & 15) == 0) {
        float M = 0.0f, pullsum = 0.0f;
        for (int i = 0; i < K_; ++i) {
            M += sValid[b * K_ + i];
            pullsum += sPull[b * K_ + i];
        }
        const float pull_b = pullsum / fmaxf(M, 1.0f);
        float push = 0.0f;
        for (int i = 0; i < K_; ++i) {
            for (int j = i + 1; j < K_; ++j) {
                if (sValid[b * K_ + i] > 0.0f && sValid[b * K_ + j] > 0.0f) {
                    float d2 = 0.0f;
                    for (int c = 0; c < C_; ++c) {
                        const float d = sMean[b * K_ + i][c] - sMean[b * K_ + j][c];
                        d2 += d * d;
                    }
                    const float dist = sqrtf(d2 + 1e-12f);
                    const float hinge = fmaxf(DELTA_PUSH - dist, 0.0f);
                    push += hinge * hinge;
                }
            }
        }
        const float npairs = M * (M - 1.0f) * 0.5f;
        const float push_b = (M > 1.0f) ? push / fmaxf(npairs, 1.0f) : 0.0f;
        sLoss[b] = DELTA_PULL * pull_b + push_b;
    }
    __syncthreads();

    if (t == 0) out[0] = (sLoss[0] + sLoss[1] + sLoss[2] + sLoss[3]) * 0.25f;
}

// ---------------------------------------------------------------------------
extern "C" void kernel_launch(void* const* d_in, const int* in_sizes, int n_in,
                              void* d_out, int out_size, void* d_ws, size_t ws_size,
                              hipStream_t stream) {
    const float* emb = (const float*)d_in[0];  // [B, C, H, W] f32
    const int* msk = (const int*)d_in[1];      // [B, K, H, W] i32

    float* ws = (float*)d_ws;
    float* sums = ws;                          // [B][K][C]
    float* sqs = ws + B_ * K_ * C_;            // [B][K][C]
    float* cnts = ws + 2 * B_ * K_ * C_;       // [B][K]
    const int nz = 2 * B_ * K_ * C_ + B_ * K_; // 2112 floats

    zero_ws_kernel<<<(nz + 255) / 256, 256, 0, stream>>>(ws, nz);
    seg_reduce_kernel<<<B_ * BLOCKS_PER_B, THREADS, 0, stream>>>(emb, msk, sums, sqs, cnts);
    finalize_kernel<<<1, 64, 0, stream>>>(sums, sqs, cnts, (float*)d_out);
}